// GCN_13683765805591
// MI455X (gfx1250) — compile-verified
//
#include <hip/hip_runtime.h>
#include <hip/hip_bf16.h>

typedef __attribute__((ext_vector_type(2))) float v2f;
typedef __attribute__((ext_vector_type(8))) float v8f;

#define N_NODES 10000
#define N_EDGES 320000

// ---------------------------------------------------------------------------
// WMMA fp32 GEMM: out[N x F] = A[N x K] * B[K x FB], FB >= F (B zero-padded
// when F is not a multiple of 16*NT). One wave computes a 16 x (16*NT) strip:
// NT accumulator tiles share one A fragment per K-step (A reuse, NT
// independent WMMA dependency chains). The grid covers FB exactly, so every
// load is in-bounds with no masking; only the final stores are guarded, so
// EXEC is all-1s through every WMMA (ISA 7.12 requirement).
// Fragment layouts per CDNA5 ISA 7.12.2 (32-bit, wave32):
//   A 16x4:  lane<16: row M=lane, VGPR{0,1}=K{0,1}; lane>=16: K{2,3}
//   B 4x16:  VGPR v: lane<16 -> K=v, col=lane; lane>=16 -> K=v+2, col=lane-16
//   C 16x16: VGPR v: lane<16 -> M=v, col=lane; lane>=16 -> M=v+8, col=lane-16
// ---------------------------------------------------------------------------
template <int F, int FB, int NT>
__global__ __launch_bounds__(32) void gemm_wmma_f32(
    const float* __restrict__ A, const float* __restrict__ B,
    float* __restrict__ out, int K)
{
    const int tileM   = blockIdx.x;
    const int lane    = threadIdx.x;
    const int half    = lane >> 4;       // 0: lanes 0-15, 1: lanes 16-31
    const int l15     = lane & 15;
    const int colBase = blockIdx.y * (16 * NT);

    const int row = tileM * 16 + l15;    // A row handled by this lane

    int colv[NT];
    #pragma unroll
    for (int t = 0; t < NT; ++t) colv[t] = colBase + t * 16 + l15;  // < FB always

    v8f acc[NT];
    #pragma unroll
    for (int t = 0; t < NT; ++t) acc[t] = (v8f){};

    const float* __restrict__ Arow = A + (long long)row * K;
    for (int k = 0; k < K; k += 4) {
        const int ka = k + half * 2;                 // lane's 2 contiguous K's
        const v2f a = *(const v2f*)(Arow + ka);

        const float* __restrict__ B0 = B + (long long)ka * FB;
        const float* __restrict__ B1 = B0 + FB;
        #pragma unroll
        for (int t = 0; t < NT; ++t) {
            v2f b;
            b[0] = B0[colv[t]];
            b[1] = B1[colv[t]];
            acc[t] = __builtin_amdgcn_wmma_f32_16x16x4_f32(
                false, a, false, b, (short)0, acc[t], false, false);
        }
    }

    #pragma unroll
    for (int t = 0; t < NT; ++t) {
        if (colv[t] < F) {                            // only ragged for F=56
            #pragma unroll
            for (int v = 0; v < 8; ++v) {
                const int m = tileM * 16 + v + half * 8;
                out[(long long)m * F + colv[t]] = acc[t][v];
            }
        }
    }
}

// ---------------------------------------------------------------------------
// Zero-pad W [K x F] -> Wp [K x FB]
// ---------------------------------------------------------------------------
template <int F, int FB>
__global__ void pad_weights(const float* __restrict__ W, float* __restrict__ Wp, int n) {
    int i = blockIdx.x * blockDim.x + threadIdx.x;     // n = K * FB
    if (i >= n) return;
    const int k = i / FB;
    const int c = i - k * FB;
    Wp[i] = (c < F) ? W[k * F + c] : 0.0f;
}

// ---------------------------------------------------------------------------
// Degree / normalization
// ---------------------------------------------------------------------------
__global__ void deg_init(float* __restrict__ deg, int n) {
    int i = blockIdx.x * blockDim.x + threadIdx.x;
    if (i < n) deg[i] = 1.0f;                      // self-loop
}
__global__ void deg_accum(const int* __restrict__ dst, float* __restrict__ deg, int e) {
    int i = blockIdx.x * blockDim.x + threadIdx.x;
    if (i < e) atomicAdd(&deg[dst[i]], 1.0f);
}
__global__ void deg_rsqrt(float* __restrict__ deg, int n) {
    int i = blockIdx.x * blockDim.x + threadIdx.x;
    if (i < n) deg[i] = rsqrtf(deg[i]);            // deg buffer becomes dinv
}

// ---------------------------------------------------------------------------
// Zero fill (float4)
// ---------------------------------------------------------------------------
__global__ void fill_zero4(float4* __restrict__ p, int n4) {
    int i = blockIdx.x * blockDim.x + threadIdx.x;
    if (i < n4) p[i] = make_float4(0.f, 0.f, 0.f, 0.f);
}

// ---------------------------------------------------------------------------
// Edge scatter: agg[dst, f..f+3] += h[src, f..f+3] * dinv[src]*dinv[dst]
// One thread per (edge, 4 features): float4 gather, 4 scalar atomics.
// ---------------------------------------------------------------------------
template <int F>
__global__ void edge_scatter(const int* __restrict__ src, const int* __restrict__ dst,
                             const float* __restrict__ h, const float* __restrict__ dinv,
                             float* __restrict__ agg)
{
    constexpr int Q = F / 4;
    long long idx = (long long)blockIdx.x * blockDim.x + threadIdx.x;
    if (idx >= (long long)N_EDGES * Q) return;
    const int e  = (int)(idx / Q);
    const int f4 = (int)(idx - (long long)e * Q) * 4;
    const int s = src[e], d = dst[e];
    const float coef = dinv[s] * dinv[d];
    const float4 hv = *(const float4*)(h + (long long)s * F + f4);
    float* ap = agg + (long long)d * F + f4;
    atomicAdd(ap + 0, hv.x * coef);
    atomicAdd(ap + 1, hv.y * coef);
    atomicAdd(ap + 2, hv.z * coef);
    atomicAdd(ap + 3, hv.w * coef);
}

// ---------------------------------------------------------------------------
// Combine: agg = leaky_relu(agg + h * dinv^2 + b, 0.1), in place (float4)
// ---------------------------------------------------------------------------
__device__ __forceinline__ float lrelu(float v) { return v >= 0.0f ? v : 0.1f * v; }

template <int F>
__global__ void combine_act(float* __restrict__ agg, const float* __restrict__ h,
                            const float* __restrict__ dinv, const float* __restrict__ b)
{
    constexpr int Q = F / 4;
    int idx = blockIdx.x * blockDim.x + threadIdx.x;
    if (idx >= N_NODES * Q) return;
    const int r  = idx / Q;
    const int f4 = (idx - r * Q) * 4;
    const float di = dinv[r];
    const float di2 = di * di;
    const long long o = (long long)r * F + f4;
    float4 av = *(float4*)(agg + o);
    const float4 hv = *(const float4*)(h + o);
    const float4 bv = *(const float4*)(b + f4);
    av.x = lrelu(av.x + hv.x * di2 + bv.x);
    av.y = lrelu(av.y + hv.y * di2 + bv.y);
    av.z = lrelu(av.z + hv.z * di2 + bv.z);
    av.w = lrelu(av.w + hv.w * di2 + bv.w);
    *(float4*)(agg + o) = av;
}

// ---------------------------------------------------------------------------
// Bias + softmax over 16 channels, one thread per row
// ---------------------------------------------------------------------------
__global__ void softmax16(const float* __restrict__ logits, const float* __restrict__ bl,
                          float* __restrict__ out) {
    int r = blockIdx.x * blockDim.x + threadIdx.x;
    if (r >= N_NODES) return;
    float buf[16], m = -3.4e38f;
    #pragma unroll
    for (int i = 0; i < 16; ++i) {
        buf[i] = logits[r * 16 + i] + bl[i];
        m = fmaxf(m, buf[i]);
    }
    float s = 0.0f;
    #pragma unroll
    for (int i = 0; i < 16; ++i) { buf[i] = expf(buf[i] - m); s += buf[i]; }
    const float inv = 1.0f / s;
    #pragma unroll
    for (int i = 0; i < 16; ++i) out[r * 16 + i] = buf[i] * inv;
}

// ---------------------------------------------------------------------------
// Host-side orchestration
// ---------------------------------------------------------------------------
static inline int cdiv(long long a, int b) { return (int)((a + b - 1) / b); }

extern "C" void kernel_launch(void* const* d_in, const int* in_sizes, int n_in,
                              void* d_out, int out_size, void* d_ws, size_t ws_size,
                              hipStream_t stream) {
    const float* x  = (const float*)d_in[0];
    const int*   ei = (const int*)d_in[1];
    const float* W1 = (const float*)d_in[2];  const float* b1 = (const float*)d_in[3];
    const float* W2 = (const float*)d_in[4];  const float* b2 = (const float*)d_in[5];
    const float* W3 = (const float*)d_in[6];  const float* b3 = (const float*)d_in[7];
    const float* W4 = (const float*)d_in[8];  const float* b4 = (const float*)d_in[9];
    const float* Wl = (const float*)d_in[10]; const float* bl = (const float*)d_in[11];
    float* outp = (float*)d_out;

    const int E = in_sizes[1] / 2;
    const int* src = ei;
    const int* dst = ei + E;

    // Workspace carve-up (floats)
    float* ws   = (float*)d_ws;
    float* dinv = ws;                                   // 10000
    float* h1   = dinv + 10000;  // keeps float4 sections 16B aligned
    float* a1   = h1 + (long long)N_NODES * 256;
    float* h2   = a1 + (long long)N_NODES * 256;
    float* a2   = h2 + (long long)N_NODES * 128;
    float* h3   = a2 + (long long)N_NODES * 128;
    float* a3   = h3 + (long long)N_NODES * 56;
    float* h4   = a3 + (long long)N_NODES * 56;
    float* a4   = h4 + (long long)N_NODES * 32;
    float* lg   = a4 + (long long)N_NODES * 32;
    float* W3p  = lg + (long long)N_NODES * 16;         // 128 x 64 padded W3

    const int TPB = 256;
    const int MT  = N_NODES / 16;   // 625 row tiles

    // --- degree normalization + padded W3 (independent prep work) ---
    deg_init <<<cdiv(N_NODES, TPB), TPB, 0, stream>>>(dinv, N_NODES);
    deg_accum<<<cdiv(E, TPB), TPB, 0, stream>>>(dst, dinv, E);
    deg_rsqrt<<<cdiv(N_NODES, TPB), TPB, 0, stream>>>(dinv, N_NODES);
    pad_weights<56, 64><<<cdiv(128 * 64, TPB), TPB, 0, stream>>>(W3, W3p, 128 * 64);

    // --- layer 1: 512 -> 256 ---
    gemm_wmma_f32<256, 256, 4><<<dim3(MT, 4), 32, 0, stream>>>(x, W1, h1, 512);
    fill_zero4<<<cdiv((long long)N_NODES * 64, TPB), TPB, 0, stream>>>((float4*)a1, N_NODES * 64);
    edge_scatter<256><<<cdiv((long long)E * 64, TPB), TPB, 0, stream>>>(src, dst, h1, dinv, a1);
    combine_act<256><<<cdiv((long long)N_NODES * 64, TPB), TPB, 0, stream>>>(a1, h1, dinv, b1);

    // --- layer 2: 256 -> 128 ---
    gemm_wmma_f32<128, 128, 4><<<dim3(MT, 2), 32, 0, stream>>>(a1, W2, h2, 256);
    fill_zero4<<<cdiv((long long)N_NODES * 32, TPB), TPB, 0, stream>>>((float4*)a2, N_NODES * 32);
    edge_scatter<128><<<cdiv((long long)E * 32, TPB), TPB, 0, stream>>>(src, dst, h2, dinv, a2);
    combine_act<128><<<cdiv((long long)N_NODES * 32, TPB), TPB, 0, stream>>>(a2, h2, dinv, b2);

    // --- layer 3: 128 -> 56 (B padded to 64 cols; stores guarded) ---
    gemm_wmma_f32<56, 64, 4><<<dim3(MT, 1), 32, 0, stream>>>(a2, W3p, h3, 128);
    fill_zero4<<<cdiv((long long)N_NODES * 14, TPB), TPB, 0, stream>>>((float4*)a3, N_NODES * 14);
    edge_scatter<56><<<cdiv((long long)E * 14, TPB), TPB, 0, stream>>>(src, dst, h3, dinv, a3);
    combine_act<56><<<cdiv((long long)N_NODES * 14, TPB), TPB, 0, stream>>>(a3, h3, dinv, b3);

    // --- layer 4: 56 -> 32 ---
    gemm_wmma_f32<32, 32, 2><<<dim3(MT, 1), 32, 0, stream>>>(a3, W4, h4, 56);
    fill_zero4<<<cdiv((long long)N_NODES * 8, TPB), TPB, 0, stream>>>((float4*)a4, N_NODES * 8);
    edge_scatter<32><<<cdiv((long long)E * 8, TPB), TPB, 0, stream>>>(src, dst, h4, dinv, a4);
    combine_act<32><<<cdiv((long long)N_NODES * 8, TPB), TPB, 0, stream>>>(a4, h4, dinv, b4);

    // --- final linear 32 -> 16, then bias + softmax ---
    gemm_wmma_f32<16, 16, 1><<<dim3(MT, 1), 32, 0, stream>>>(a4, Wl, lg, 32);
    softmax16<<<cdiv(N_NODES, TPB), TPB, 0, stream>>>(lg, bl, outp);

    (void)ws_size; (void)out_size; (void)n_in;
}